// TacticalGAT_37031208026767
// MI455X (gfx1250) — compile-verified
//
#include <hip/hip_runtime.h>
#include <hip/hip_bf16.h>
#include <math.h>

// ---------------------------------------------------------------------------
// GATv2 (2 layers) + mean pool + MLP head for MI455X (gfx1250, wave32).
// - Dense node transforms: V_WMMA_F32_16X16X4_F32 (fp32-exact matrix core).
// - Edge passes: one wave32 per edge, float4/float2 coalesced row slices,
//   __shfl_xor per-head reductions, global float atomics for segment ops.
// ---------------------------------------------------------------------------

typedef float v2f __attribute__((ext_vector_type(2)));
typedef float v8f __attribute__((ext_vector_type(8)));

#define H1C 4
#define C1C 32
#define HC1 128   // H1*C1
#define C2C 64
#define FINC 128
#define GCLS 10

// ---------------- generic fill ----------------
__global__ void fill_kernel(float* __restrict__ p, float v, long n) {
  long i = (long)blockIdx.x * blockDim.x + threadIdx.x;
  if (i < n) p[i] = v;
}

// ---------------- float atomic max (CAS) ----------------
__device__ inline void atomicMaxF(float* addr, float val) {
  int* ai = (int*)addr;
  int cur = __float_as_int(*addr);
  while (__int_as_float(cur) < val) {
    int old = atomicCAS(ai, cur, __float_as_int(val));
    if (old == cur) break;
    cur = old;
  }
}

// ---------------- self loop mean edge-attr ----------------
__global__ void selfloop_accum(const int* __restrict__ ei, const float* __restrict__ eattr,
                               float* __restrict__ loopattr, float* __restrict__ cnt, int E) {
  int e = blockIdx.x * blockDim.x + threadIdx.x;
  if (e >= E) return;
  int d = ei[E + e];
  atomicAdd(&cnt[d], 1.0f);
  atomicAdd(&loopattr[d * 3 + 0], eattr[e * 3 + 0]);
  atomicAdd(&loopattr[d * 3 + 1], eattr[e * 3 + 1]);
  atomicAdd(&loopattr[d * 3 + 2], eattr[e * 3 + 2]);
}

__global__ void selfloop_final(float* __restrict__ loopattr, const float* __restrict__ cnt, int N) {
  int i = blockIdx.x * blockDim.x + threadIdx.x;
  if (i >= N) return;
  float c = fmaxf(cnt[i], 1.0f);
  loopattr[i * 3 + 0] /= c;
  loopattr[i * 3 + 1] /= c;
  loopattr[i * 3 + 2] /= c;
}

// ---------------- fp32 WMMA GEMM: Y[M,Nout] = X[M,K] @ W[Nout,K]^T + bias ----
// One wave (32 lanes) computes one 16x16 output tile via V_WMMA_F32_16X16X4_F32.
// A 16x4 layout : lane L holds row m=L%16, K pair koff=(L/16)*2 .. +1
// B 4x16 layout : lane L holds col n=L%16, same K pair (B[k][n] = W[n][k])
// C/D 16x16     : vgpr r holds row m = r + (L/16)*8, col n = L%16
__global__ void gemm_wmma_f32(const float* __restrict__ X, const float* __restrict__ W,
                              const float* __restrict__ bias, float* __restrict__ Y,
                              int M, int K, int Nout) {
  const int lane = threadIdx.x & 31;
  const int mn   = lane & 15;
  const int koff = (lane >> 4) << 1;
  const int mbase = blockIdx.x * 16;
  const int nbase = blockIdx.y * 16;

  int mrow = mbase + mn;
  int mclamp = mrow < M ? mrow : (M - 1);
  const float* xrow = X + (long)mclamp * K;
  const float* wrow = W + (long)(nbase + mn) * K;

  v8f c = {};
  for (int k0 = 0; k0 < K; k0 += 4) {
    v2f a, b;
    a.x = xrow[k0 + koff];
    a.y = xrow[k0 + koff + 1];
    b.x = wrow[k0 + koff];
    b.y = wrow[k0 + koff + 1];
    c = __builtin_amdgcn_wmma_f32_16x16x4_f32(false, a, false, b, (short)0, c, false, false);
  }

  const float bcol = bias[nbase + mn];
  const int mhi = (lane >> 4) * 8;
#pragma unroll
  for (int r = 0; r < 8; ++r) {
    int row = mbase + mhi + r;
    if (row < M) Y[(long)row * Nout + nbase + mn] = c[r] + bcol;
  }
}

// ---------------- edge pass A: score + segment max (wave per edge) ---------
// edges e < E come from edge_index; e >= E are self loops (src=dst=e-E).
// Each lane owns VEC = H*C/32 consecutive channels; per-head partial dot
// products are reduced across the GRP = C/VEC lanes of that head.
template<int H, int C>
__global__ void edge_score_wave(const int* __restrict__ ei, const float* __restrict__ xl,
                                const float* __restrict__ xr, const float* __restrict__ eattr,
                                const float* __restrict__ loopattr, const float* __restrict__ We,
                                const float* __restrict__ att, float* __restrict__ scores,
                                float* __restrict__ smax, int E, int N) {
  constexpr int HC  = H * C;
  constexpr int VEC = HC / 32;   // channels per lane
  constexpr int GRP = C / VEC;   // lanes per head
  int e    = (int)((blockIdx.x * (long)blockDim.x + threadIdx.x) >> 5);
  int lane = threadIdx.x & 31;
  int total = E + N;
  if (e >= total) return;

  int s, d; float ea0, ea1, ea2;
  if (e < E) {
    s = ei[e]; d = ei[E + e];
    ea0 = eattr[e * 3 + 0]; ea1 = eattr[e * 3 + 1]; ea2 = eattr[e * 3 + 2];
  } else {
    s = d = e - E;
    ea0 = loopattr[s * 3 + 0]; ea1 = loopattr[s * 3 + 1]; ea2 = loopattr[s * 3 + 2];
  }

  const int base = lane * VEC;
  const float* xls = xl + (long)s * HC + base;   // contiguous slice -> b128/b64 load
  const float* xrd = xr + (long)d * HC + base;
  float part = 0.0f;
#pragma unroll
  for (int j = 0; j < VEC; ++j) {
    int oc = base + j;
    float ee = ea0 * We[oc * 3 + 0] + ea1 * We[oc * 3 + 1] + ea2 * We[oc * 3 + 2];
    float m = xls[j] + xrd[j] + ee;
    m = m > 0.0f ? m : 0.2f * m;                 // leaky_relu(0.2)
    part += m * att[oc];
  }
  // reduce within each head's lane group
#pragma unroll
  for (int off = 1; off < GRP; off <<= 1)
    part += __shfl_xor(part, off, 32);
  if ((lane % GRP) == 0) {
    int h = base / C;
    scores[(long)e * H + h] = part;
    atomicMaxF(&smax[d * H + h], part);
  }
}

// ---------------- edge pass B: exp + segment sum (thread per edge-head) ----
__global__ void edge_exp(const int* __restrict__ ei, float* __restrict__ scores,
                         const float* __restrict__ smax, float* __restrict__ denom,
                         int E, int N, int H) {
  int e = blockIdx.x * blockDim.x + threadIdx.x;
  int h = blockIdx.y;
  int total = E + N;
  if (e >= total) return;
  int d = (e < E) ? ei[E + e] : (e - E);
  float ex = expf(scores[(long)e * H + h] - smax[d * H + h]);
  scores[(long)e * H + h] = ex;
  atomicAdd(&denom[d * H + h], ex);
}

// ---------------- edge pass C: alpha-weighted scatter (wave per edge) ------
template<int H, int C>
__global__ void edge_scatter_wave(const int* __restrict__ ei, const float* __restrict__ scores,
                                  const float* __restrict__ denom, const float* __restrict__ xl,
                                  float* __restrict__ out, int E, int N) {
  constexpr int HC  = H * C;
  constexpr int VEC = HC / 32;
  int e    = (int)((blockIdx.x * (long)blockDim.x + threadIdx.x) >> 5);
  int lane = threadIdx.x & 31;
  int total = E + N;
  if (e >= total) return;

  int s, d;
  if (e < E) { s = ei[e]; d = ei[E + e]; } else { s = d = e - E; }

  const int base = lane * VEC;
  const int h = base / C;
  float alpha = scores[(long)e * H + h] / denom[d * H + h];
  const float* xls = xl + (long)s * HC + base;
  float* outd = out + (long)d * HC + base;
#pragma unroll
  for (int j = 0; j < VEC; ++j)
    atomicAdd(&outd[j], alpha * xls[j]);         // line-local across the wave
}

// ---------------- bias + ELU (in place) ------------------------------------
__global__ void bias_elu(float* __restrict__ y, const float* __restrict__ b, int N, int F) {
  long i = (long)blockIdx.x * blockDim.x + threadIdx.x;
  if (i >= (long)N * F) return;
  int c = (int)(i % F);
  float v = y[i] + b[c];
  y[i] = v > 0.0f ? v : (expf(v) - 1.0f);   // elu
}

// ---------------- pooling --------------------------------------------------
__global__ void pool_accum(const float* __restrict__ h2, const int* __restrict__ batch,
                           float* __restrict__ pooled, float* __restrict__ gcnt, int N) {
  // wave per node: 32 lanes cover 64 channels as float2 slices
  int i    = (int)((blockIdx.x * (long)blockDim.x + threadIdx.x) >> 5);
  int lane = threadIdx.x & 31;
  if (i >= N) return;
  int g = batch[i];
  if (lane == 0) atomicAdd(&gcnt[g], 1.0f);
  const float* row = h2 + (long)i * C2C + lane * 2;
  float* dst = pooled + (long)g * C2C + lane * 2;
  atomicAdd(&dst[0], row[0]);
  atomicAdd(&dst[1], row[1]);
}

__global__ void pool_final(float* __restrict__ pooled, const float* __restrict__ gcnt, int G) {
  int i = blockIdx.x * blockDim.x + threadIdx.x;
  if (i >= G * C2C) return;
  int g = i / C2C;
  pooled[i] /= fmaxf(gcnt[g], 1.0f);
}

// ---------------- MLP head -------------------------------------------------
__global__ void mlp_lin1(const float* __restrict__ pooled, const float* __restrict__ u,
                         const float* __restrict__ W, const float* __restrict__ b,
                         float* __restrict__ z, int G) {
  int t = blockIdx.x * blockDim.x + threadIdx.x;
  if (t >= G * 32) return;
  int g = t / 32, j = t % 32;
  float acc = b[j];
  for (int c = 0; c < C2C; ++c) acc += pooled[g * C2C + c] * W[j * 65 + c];
  acc += u[g] * W[j * 65 + 64];
  z[t] = acc > 0.0f ? acc : 0.0f;          // relu
}

__global__ void mlp_head(const float* __restrict__ z, const float* __restrict__ W,
                         const float* __restrict__ b, float* __restrict__ out, int G) {
  int t = blockIdx.x * blockDim.x + threadIdx.x;
  if (t >= G * GCLS) return;
  int g = t / GCLS, k = t % GCLS;
  float acc = b[k];
  for (int j = 0; j < 32; ++j) acc += z[g * 32 + j] * W[k * 32 + j];
  out[t] = acc;
}

// ---------------------------------------------------------------------------
extern "C" void kernel_launch(void* const* d_in, const int* in_sizes, int n_in,
                              void* d_out, int out_size, void* d_ws, size_t ws_size,
                              hipStream_t stream) {
  const float* x      = (const float*)d_in[0];
  const int*   ei     = (const int*)  d_in[1];
  const float* eattr  = (const float*)d_in[2];
  const int*   batch  = (const int*)  d_in[3];
  const float* u      = (const float*)d_in[4];
  const float* Wl1    = (const float*)d_in[5];
  const float* bl1    = (const float*)d_in[6];
  const float* Wr1    = (const float*)d_in[7];
  const float* br1    = (const float*)d_in[8];
  const float* We1    = (const float*)d_in[9];
  const float* att1   = (const float*)d_in[10];
  const float* b1     = (const float*)d_in[11];
  const float* Wl2    = (const float*)d_in[12];
  const float* bl2    = (const float*)d_in[13];
  const float* Wr2    = (const float*)d_in[14];
  const float* br2    = (const float*)d_in[15];
  const float* We2    = (const float*)d_in[16];
  const float* att2   = (const float*)d_in[17];
  const float* b2     = (const float*)d_in[18];
  const float* W_lin1 = (const float*)d_in[19];
  const float* b_lin1 = (const float*)d_in[20];
  const float* W_head = (const float*)d_in[21];
  const float* b_head = (const float*)d_in[22];

  const int N  = in_sizes[0] / FINC;
  const int E  = in_sizes[1] / 2;
  const int G  = in_sizes[4];
  const int EP = E + N;

  // ---- workspace layout (floats) ----
  float* ws = (float*)d_ws;
  long off = 0;
  float* loopattr = ws + off; off += (long)N * 3;
  float* cnt      = ws + off; off += N;
  float* xl1      = ws + off; off += (long)N * HC1;   // reused as xl2
  float* xr1      = ws + off; off += (long)N * HC1;   // reused as xr2
  float* h1       = ws + off; off += (long)N * HC1;   // layer1 accum -> elu output
  float* scores   = ws + off; off += (long)EP * H1C;  // reused by layer2 (H=1)
  float* smax     = ws + off; off += (long)N * H1C;
  float* denom    = ws + off; off += (long)N * H1C;
  float* h2       = ws + off; off += (long)N * C2C;   // layer2 accum -> elu output
  float* pooled   = ws + off; off += (long)G * C2C;
  float* gcnt     = ws + off; off += G;
  float* z        = ws + off; off += (long)G * 32;
  (void)ws_size; (void)n_in; (void)out_size;

  const int TB = 256;            // 8 waves per block
  const int WPB = TB / 32;
  const float NEG_INF = -__builtin_huge_valf();
  #define CEIL(a, b) (((a) + (b) - 1) / (b))

  // ---- self loops: mean incoming edge attr ----
  fill_kernel<<<CEIL((long)N * 4, TB), TB, 0, stream>>>(loopattr, 0.0f, (long)N * 4); // loopattr+cnt contiguous
  selfloop_accum<<<CEIL(E, TB), TB, 0, stream>>>(ei, eattr, loopattr, cnt, E);
  selfloop_final<<<CEIL(N, TB), TB, 0, stream>>>(loopattr, cnt, N);

  // ---- layer 1 node transforms (WMMA fp32) ----
  dim3 g1(CEIL(N, 16), HC1 / 16);
  gemm_wmma_f32<<<g1, 32, 0, stream>>>(x, Wl1, bl1, xl1, N, FINC, HC1);
  gemm_wmma_f32<<<g1, 32, 0, stream>>>(x, Wr1, br1, xr1, N, FINC, HC1);

  // ---- layer 1 attention ----
  fill_kernel<<<CEIL((long)N * H1C, TB), TB, 0, stream>>>(smax, NEG_INF, (long)N * H1C);
  fill_kernel<<<CEIL((long)N * H1C, TB), TB, 0, stream>>>(denom, 0.0f, (long)N * H1C);
  fill_kernel<<<CEIL((long)N * HC1, TB), TB, 0, stream>>>(h1, 0.0f, (long)N * HC1);
  edge_score_wave<H1C, C1C><<<CEIL(EP, WPB), TB, 0, stream>>>(ei, xl1, xr1, eattr, loopattr,
                                                              We1, att1, scores, smax, E, N);
  dim3 ge1(CEIL(EP, TB), H1C);
  edge_exp<<<ge1, TB, 0, stream>>>(ei, scores, smax, denom, E, N, H1C);
  edge_scatter_wave<H1C, C1C><<<CEIL(EP, WPB), TB, 0, stream>>>(ei, scores, denom, xl1, h1, E, N);
  bias_elu<<<CEIL((long)N * HC1, TB), TB, 0, stream>>>(h1, b1, N, HC1);

  // ---- layer 2 node transforms (WMMA fp32), buffers reused ----
  float* xl2 = xl1;
  float* xr2 = xr1;
  dim3 g2(CEIL(N, 16), C2C / 16);
  gemm_wmma_f32<<<g2, 32, 0, stream>>>(h1, Wl2, bl2, xl2, N, HC1, C2C);
  gemm_wmma_f32<<<g2, 32, 0, stream>>>(h1, Wr2, br2, xr2, N, HC1, C2C);

  // ---- layer 2 attention (H=1, C=64) ----
  fill_kernel<<<CEIL(N, TB), TB, 0, stream>>>(smax, NEG_INF, N);
  fill_kernel<<<CEIL(N, TB), TB, 0, stream>>>(denom, 0.0f, N);
  fill_kernel<<<CEIL((long)N * C2C, TB), TB, 0, stream>>>(h2, 0.0f, (long)N * C2C);
  edge_score_wave<1, C2C><<<CEIL(EP, WPB), TB, 0, stream>>>(ei, xl2, xr2, eattr, loopattr,
                                                            We2, att2, scores, smax, E, N);
  dim3 ge2(CEIL(EP, TB), 1);
  edge_exp<<<ge2, TB, 0, stream>>>(ei, scores, smax, denom, E, N, 1);
  edge_scatter_wave<1, C2C><<<CEIL(EP, WPB), TB, 0, stream>>>(ei, scores, denom, xl2, h2, E, N);
  bias_elu<<<CEIL((long)N * C2C, TB), TB, 0, stream>>>(h2, b2, N, C2C);

  // ---- mean pool per graph ----
  fill_kernel<<<CEIL((long)G * C2C + G, TB), TB, 0, stream>>>(pooled, 0.0f, (long)G * C2C + G); // pooled+gcnt contiguous
  pool_accum<<<CEIL(N, WPB), TB, 0, stream>>>(h2, batch, pooled, gcnt, N);
  pool_final<<<CEIL(G * C2C, TB), TB, 0, stream>>>(pooled, gcnt, G);

  // ---- MLP head ----
  mlp_lin1<<<CEIL(G * 32, TB), TB, 0, stream>>>(pooled, u, W_lin1, b_lin1, z, G);
  mlp_head<<<CEIL(G * GCLS, TB), TB, 0, stream>>>(z, W_head, b_head, (float*)d_out, G);
}